// MoBoAligner_30554397343928
// MI455X (gfx1250) — compile-verified
//
#include <hip/hip_runtime.h>
#include <math.h>

#define NB 4
#define NI 48
#define NJ 384
#define DT 512
#define DM 80
#define DA 128
#define LOG_EPS (-1000.0f)
#define NWAVES (NJ / 32)

typedef float v2f __attribute__((ext_vector_type(2)));
typedef float v8f __attribute__((ext_vector_type(8)));

__device__ __forceinline__ float logadd(float a, float b) {
    float mx = fmaxf(a, b);
    if (!(mx > -INFINITY)) return -INFINITY;
    float mn = fminf(a, b);
    return mx + log1pf(expf(mn - mx));
}

// inclusive log-sum-exp scan over NJ elements with NJ threads (in LDS)
__device__ __forceinline__ void lse_scan(float* s, int tid) {
    for (int off = 1; off < NJ; off <<= 1) {
        float a = s[tid];
        float b = (tid >= off) ? s[tid - off] : -INFINITY;
        __syncthreads();
        s[tid] = logadd(a, b);
        __syncthreads();
    }
}

// wave32 log-sum-exp butterfly reduction (no barriers)
__device__ __forceinline__ float wave_lse(float v) {
    for (int o = 16; o > 0; o >>= 1) v = logadd(v, __shfl_xor(v, o, 32));
    return v;
}

// ---------------------------------------------------------------- lens
__global__ void k_lens(const unsigned char* tmask, const unsigned char* mmask, int* lens) {
    int t = threadIdx.x;
    if (t < NB) {
        int c = 0;
        for (int i = 0; i < NI; ++i) c += (tmask[t * NI + i] != 0);
        lens[t] = c;
    } else if (t < 2 * NB) {
        int b = t - NB;
        int c = 0;
        for (int j = 0; j < NJ; ++j) c += (mmask[b * NJ + j] != 0);
        lens[NB + b] = c;
    }
}

// ---------------------------------------------------------------- generic f32 WMMA GEMM
// C[16x16 tile] = A(MxK,row-major,lda) @ B(KxN,row-major,ldb) (+bias[n])
__global__ void k_gemm_wmma(const float* __restrict__ A, int lda,
                            const float* __restrict__ Bm, int ldb,
                            float* __restrict__ C, int ldc, int K,
                            const float* __restrict__ bias) {
    int tm = blockIdx.x * 16, tn = blockIdx.y * 16;
    int lane = threadIdx.x;
    int mr = lane & 15;
    int kb = (lane >> 4) << 1;   // 0 for lanes 0-15, 2 for lanes 16-31
    v8f acc = {0.f, 0.f, 0.f, 0.f, 0.f, 0.f, 0.f, 0.f};
    const float* Ap = A + (size_t)tm * lda;
    const float* Bp = Bm + tn;
    for (int k = 0; k < K; k += 4) {
        if (k + 8 < K) {   // prefetch next K-slab (lowers to global_prefetch_b8)
            __builtin_prefetch(&Ap[(size_t)mr * lda + k + 8 + kb], 0, 1);
            __builtin_prefetch(&Bp[(size_t)(k + 8 + kb) * ldb + mr], 0, 1);
        }
        v2f a, b;
        a.x = Ap[(size_t)mr * lda + k + kb];
        a.y = Ap[(size_t)mr * lda + k + kb + 1];
        b.x = Bp[(size_t)(k + kb) * ldb + mr];
        b.y = Bp[(size_t)(k + kb + 1) * ldb + mr];
        acc = __builtin_amdgcn_wmma_f32_16x16x4_f32(false, a, false, b, (short)0, acc,
                                                    false, false);
    }
    int rbase = (lane < 16) ? 0 : 8;
    float bv = bias ? bias[tn + mr] : 0.0f;
    for (int r = 0; r < 8; ++r) {
        C[(size_t)(tm + rbase + r) * ldc + tn + mr] = acc[r] + bv;
    }
}

// ---------------------------------------------------------------- energy + suffix-LSE T + Z
__global__ void k_energy(const float* __restrict__ pt, const float* __restrict__ pm,
                         const float* __restrict__ noise, const float* __restrict__ vw,
                         const float* __restrict__ vb, const unsigned char* tmask,
                         const unsigned char* mmask, const int* lens,
                         float* __restrict__ mArr, float* __restrict__ Tarr,
                         float* __restrict__ Zarr) {
    int bi = blockIdx.x;
    int b = bi / NI, i = bi % NI;
    int j = threadIdx.x;
    __shared__ float ptS[DA];
    __shared__ float buf[NJ];
    if (j < DA) ptS[j] = pt[(size_t)bi * DA + j];
    __syncthreads();

    const float* pmr = pm + (size_t)(b * NJ + j) * DA;
    float s = 0.f;
    for (int d = 0; d < DA; ++d) s += vw[d] * tanhf(ptS[d] + pmr[d]);
    float e = s + vb[0] + 2.0f * noise[(size_t)bi * NJ + j];

    int tl = lens[b], ml = lens[NB + b];
    bool jm = (tmask[b * NI + i] != 0) && (mmask[b * NJ + j] != 0) &&
              !((i == tl - 1) && (j != ml - 1));
    float mv = jm ? e : LOG_EPS;
    mArr[(size_t)bi * NJ + j] = mv;

    buf[NJ - 1 - j] = mv;   // reversed -> prefix scan == suffix LSE
    __syncthreads();
    lse_scan(buf, j);
    float Tj = buf[NJ - 1 - j];           // LSE_{j' >= j} m
    Tarr[(size_t)bi * NJ + j] = Tj;

    bool kvalid = (mmask[b * NJ + j] != 0);
    float Zk;
    if (kvalid) {
        float pad = (j > 0) ? (LOG_EPS + logf((float)j)) : -INFINITY;
        Zk = logadd(Tj, pad);
    } else {
        Zk = LOG_EPS + logf((float)NJ);
    }
    Zarr[(size_t)bi * NJ + j] = Zk;
}

// ---------------------------------------------------------------- banded forward DP (log)
__global__ void k_dp(const float* __restrict__ mArr, const float* __restrict__ Zarr,
                     const unsigned char* mmask, float* __restrict__ Bij) {
    int b = blockIdx.x;
    int tid = threadIdx.x;   // == k == c
    __shared__ float g[NJ], sg[NJ];
    __shared__ float red[NWAVES];
    bool kvalid = (mmask[b * NJ + tid] != 0);
    float pv = (tid == 0) ? 0.0f : -INFINITY;   // prev row (own index only)
    Bij[(size_t)(b * NI + 0) * NJ + tid] = pv;

    for (int i = 1; i < NI; ++i) {
        float Zk = Zarr[(size_t)(b * NI + i - 1) * NJ + tid];
        float base = pv - Zk;
        g[tid] = kvalid ? base : -INFINITY;
        sg[NJ - 1 - tid] = kvalid ? base : -INFINITY;
        // totH = LSE over invalid k of (prev - Z): wave butterfly + 12-partial combine
        float hw = wave_lse(kvalid ? -INFINITY : base);
        if ((tid & 31) == 0) red[tid >> 5] = hw;
        __syncthreads();
        lse_scan(g, tid);     // prefix LSE of valid prev-Z
        lse_scan(sg, tid);    // suffix LSE (reversed)
        if (tid < 32) {
            float t = (tid < NWAVES) ? red[tid] : -INFINITY;
            t = wave_lse(t);
            if (tid == 0) red[0] = t;
        }
        __syncthreads();
        float totH = red[0];

        int c = tid;
        float val = -INFINITY;
        if (c >= i && c <= (NJ - NI + i + 1)) {
            int jp = c - 1;   // 0..NJ-1
            float t1 = mArr[(size_t)(b * NI + i - 1) * NJ + jp] + g[jp];
            float sGnext = (jp + 1 < NJ) ? sg[NJ - 1 - (jp + 1)] : -INFINITY;
            float t2 = LOG_EPS + logadd(sGnext, totH);
            val = logadd(t1, t2);
        }
        __syncthreads();
        pv = val;
        Bij[(size_t)(b * NI + i) * NJ + tid] = val;
    }
}

// ---------------------------------------------------------------- soft = LSE_k(Bij + geq)
__global__ void k_soft(const float* __restrict__ Bij, const float* __restrict__ Zarr,
                       const float* __restrict__ Tarr, const unsigned char* tmask,
                       const unsigned char* mmask, const int* lens,
                       float* __restrict__ soft, float* __restrict__ P) {
    int bi = blockIdx.x;
    int b = bi / NI, i = bi % NI;
    int tid = threadIdx.x;   // == k == j
    __shared__ float g[NJ], u[NJ], vv[NJ];
    float Bk = Bij[(size_t)bi * NJ + tid];
    float Zk = Zarr[(size_t)bi * NJ + tid];
    bool kvalid = (mmask[b * NJ + tid] != 0);
    g[tid] = kvalid ? (Bk - Zk) : -INFINITY;
    u[NJ - 1 - tid] = Bk;
    vv[tid] = kvalid ? -INFINITY : Bk;
    __syncthreads();
    lse_scan(g, tid);
    lse_scan(u, tid);
    lse_scan(vv, tid);

    int j = tid;
    float totU = u[NJ - 1];
    float sUnext = (j + 1 < NJ) ? u[NJ - 1 - (j + 1)] : -INFINITY;
    int tl = lens[b], ml = lens[NB + b];
    bool jm = (tmask[b * NI + i] != 0) && (mmask[b * NJ + j] != 0) &&
              !((i == tl - 1) && (j != ml - 1));
    float t1 = jm ? (Tarr[(size_t)bi * NJ + j] + g[j]) : -INFINITY;
    float t2 = LOG_EPS + (jm ? logadd(sUnext, vv[j]) : totU);
    float s = logadd(t1, t2);
    soft[(size_t)bi * NJ + j] = s;
    P[(size_t)bi * NJ + j] = expf(s);
}

// ---------------------------------------------------------------- expanded = exp(soft)^T @ text
__global__ void k_expand_wmma(const float* __restrict__ P, const float* __restrict__ text,
                              float* __restrict__ out) {
    int b = blockIdx.z;
    int tj = blockIdx.x * 16;   // row block over J
    int tn = blockIdx.y * 16;   // col block over DT
    int lane = threadIdx.x;
    int mr = lane & 15;
    int kb = (lane >> 4) << 1;
    const float* Pb = P + (size_t)b * NI * NJ;
    const float* Tb = text + (size_t)b * NI * DT;
    v8f acc = {0.f, 0.f, 0.f, 0.f, 0.f, 0.f, 0.f, 0.f};
    for (int k = 0; k < NI; k += 4) {
        if (k + 8 < NI) {
            __builtin_prefetch(&Pb[(size_t)(k + 8 + kb) * NJ + tj + mr], 0, 1);
            __builtin_prefetch(&Tb[(size_t)(k + 8 + kb) * DT + tn + mr], 0, 1);
        }
        v2f a, bv;
        a.x = Pb[(size_t)(k + kb) * NJ + tj + mr];     // A[j,i] = P[b,i,j] (transposed read)
        a.y = Pb[(size_t)(k + kb + 1) * NJ + tj + mr];
        bv.x = Tb[(size_t)(k + kb) * DT + tn + mr];
        bv.y = Tb[(size_t)(k + kb + 1) * DT + tn + mr];
        acc = __builtin_amdgcn_wmma_f32_16x16x4_f32(false, a, false, bv, (short)0, acc,
                                                    false, false);
    }
    int rbase = (lane < 16) ? 0 : 8;
    for (int r = 0; r < 8; ++r) {
        out[(size_t)b * NJ * DT + (size_t)(tj + rbase + r) * DT + tn + mr] = acc[r];
    }
}

// ----------------------------------------------------------------
extern "C" void kernel_launch(void* const* d_in, const int* in_sizes, int n_in,
                              void* d_out, int out_size, void* d_ws, size_t ws_size,
                              hipStream_t stream) {
    const float* text  = (const float*)d_in[0];
    const float* mel   = (const float*)d_in[1];
    const float* noise = (const float*)d_in[2];
    const float* Wt    = (const float*)d_in[3];
    const float* Wm    = (const float*)d_in[4];
    const float* bm    = (const float*)d_in[5];
    const float* vw    = (const float*)d_in[6];
    const float* vb    = (const float*)d_in[7];
    const unsigned char* tmask = (const unsigned char*)d_in[8];
    const unsigned char* mmask = (const unsigned char*)d_in[9];

    float* out = (float*)d_out;
    float* soft = out;                              // NB*NI*NJ
    float* expanded = out + (size_t)NB * NI * NJ;   // NB*NJ*DT

    float* ws = (float*)d_ws;
    size_t o = 0;
    float* pt   = ws + o; o += (size_t)NB * NI * DA;
    float* pm   = ws + o; o += (size_t)NB * NJ * DA;
    float* mArr = ws + o; o += (size_t)NB * NI * NJ;
    float* Tarr = ws + o; o += (size_t)NB * NI * NJ;
    float* Zarr = ws + o; o += (size_t)NB * NI * NJ;
    float* Bij  = ws + o; o += (size_t)NB * NI * NJ;
    float* P    = ws + o; o += (size_t)NB * NI * NJ;
    int* lens   = (int*)(ws + o);

    k_lens<<<1, 32, 0, stream>>>(tmask, mmask, lens);
    // pt = text @ W_text      (M=192,N=128,K=512)
    k_gemm_wmma<<<dim3((NB * NI) / 16, DA / 16), 32, 0, stream>>>(text, DT, Wt, DA, pt, DA, DT,
                                                                  nullptr);
    // pm = mel @ W_mel + b    (M=1536,N=128,K=80)
    k_gemm_wmma<<<dim3((NB * NJ) / 16, DA / 16), 32, 0, stream>>>(mel, DM, Wm, DA, pm, DA, DM,
                                                                  bm);
    k_energy<<<NB * NI, NJ, 0, stream>>>(pt, pm, noise, vw, vb, tmask, mmask, lens, mArr, Tarr,
                                         Zarr);
    k_dp<<<NB, NJ, 0, stream>>>(mArr, Zarr, mmask, Bij);
    k_soft<<<NB * NI, NJ, 0, stream>>>(Bij, Zarr, Tarr, tmask, mmask, lens, soft, P);
    // expanded[b] = P[b]^T (J x I) @ text[b] (I x D_TEXT)
    k_expand_wmma<<<dim3(NJ / 16, DT / 16, NB), 32, 0, stream>>>(P, text, expanded);
}